// PiecewiseCubicCDF_91319594647693
// MI455X (gfx1250) — compile-verified
//
#include <hip/hip_runtime.h>
#include <hip/hip_bf16.h>
#include <math.h>

#define Dfeat 1024
#define NBATCH 4096
#define KBINS 32
#define FT 32            // features per block tile
#define ST 33            // padded LDS row stride (bank-conflict free: gcd(33,64)=1)
#define THREADS 512
#define ROWCHUNKS 32
#define MINW 0.001f
#define MINH 0.001f

typedef __attribute__((ext_vector_type(2))) float v2f;
typedef __attribute__((ext_vector_type(8))) float v8f;

__global__ void zero_lad_kernel(float* __restrict__ lad) {
    int i = blockIdx.x * blockDim.x + threadIdx.x;
    if (i < NBATCH) lad[i] = 0.0f;
}

__global__ __launch_bounds__(THREADS)
void spline_fused_kernel(const float* __restrict__ x,
                         const float* __restrict__ uw,
                         const float* __restrict__ uh,
                         const float* __restrict__ udl,
                         const float* __restrict__ udr,
                         float* __restrict__ out,    // B*D
                         float* __restrict__ lad)    // B (pre-zeroed)
{
    __shared__ float wS[FT * ST];     // widths
    __shared__ float hS[FT * ST];     // heights
    __shared__ float locsS[FT * ST];  // 33 search locations per feature
    __shared__ float aS[FT * ST];
    __shared__ float bS[FT * ST];
    __shared__ float cS[FT * ST];     // derivatives[0..31] == c
    __shared__ float dS[FT * ST];     // padded cumheights (left heights)

    const int t     = threadIdx.x;
    const int lane  = t & 31;
    const int wave  = t >> 5;
    const int fbase = blockIdx.x * FT;

    // ---- Stage raw logits (coalesced) ----
    for (int i = t; i < FT * KBINS; i += THREADS) {
        int fl = i >> 5, j = i & 31;
        wS[fl * ST + j] = uw[(fbase + fl) * KBINS + j];
        hS[fl * ST + j] = uh[(fbase + fl) * KBINS + j];
    }
    __syncthreads();

    // ---- Softmax + min-width/height affine, one thread per feature ----
    if (t < FT) {
        float* row = &wS[t * ST];
        float mx = row[0];
        for (int j = 1; j < KBINS; ++j) mx = fmaxf(mx, row[j]);
        float sum = 0.0f;
        for (int j = 0; j < KBINS; ++j) { float e = expf(row[j] - mx); row[j] = e; sum += e; }
        float sc = (1.0f - MINW * KBINS) / sum;
        for (int j = 0; j < KBINS; ++j) row[j] = MINW + row[j] * sc;

        row = &hS[t * ST];
        mx = row[0];
        for (int j = 1; j < KBINS; ++j) mx = fmaxf(mx, row[j]);
        sum = 0.0f;
        for (int j = 0; j < KBINS; ++j) { float e = expf(row[j] - mx); row[j] = e; sum += e; }
        sc = (1.0f - MINH * KBINS) / sum;
        for (int j = 0; j < KBINS; ++j) row[j] = MINH + row[j] * sc;
    }
    __syncthreads();

    // ---- Prefix sums via triangular GEMM on the matrix core ----
    // cum[f][n] = sum_k w[f][k] * U[k][n], U[k][n] = (k<=n), exact f32 WMMA.
    // waves 0..1 each own a 16-feature tile; EXEC is all-ones (uniform branch).
    if (wave < 2) {
        const int tb = wave * 16;          // local feature tile base
        const int m  = lane & 15;
        const int hh = lane >> 4;
        for (int mat = 0; mat < 2; ++mat) {
            const float* src = mat ? hS : wS;
            float*       dst = mat ? dS : locsS;   // cumheights -> dS[.][bin+1], cumwidths -> locs[.][bin+1]
            for (int ntile = 0; ntile < 2; ++ntile) {
                const int n = ntile * 16 + m;      // output bin column for this lane
                v8f acc = {};
#pragma unroll
                for (int k0 = 0; k0 < 32; k0 += 4) {
                    // A 16x4 f32 layout: lane m holds (M=m, K=k0+2h) and (M=m, K=k0+2h+1)
                    v2f a, b;
                    const int ka = k0 + 2 * hh;
                    a.x = src[(tb + m) * ST + ka];
                    a.y = src[(tb + m) * ST + ka + 1];
                    // B 4x16 fragment generated in registers from (k,n)
                    b.x = (ka     <= n) ? 1.0f : 0.0f;
                    b.y = (ka + 1 <= n) ? 1.0f : 0.0f;
                    acc = __builtin_amdgcn_wmma_f32_16x16x4_f32(
                        false, a, false, b, (short)0, acc, false, false);
                }
                // C/D layout: VGPR r -> (M = r + 8*half, N = lane&15)
#pragma unroll
                for (int r = 0; r < 8; ++r) {
                    int feat = tb + r + 8 * hh;
                    dst[feat * ST + n + 1] = acc[r];
                }
            }
        }
    }
    __syncthreads();

    // ---- Finish per-feature coefficients ----
    if (t < FT) {
        const int fg = fbase + t;
        float* wrow = &wS[t * ST];
        float* hrow = &hS[t * ST];
        locsS[t * ST + 0]  = 0.0f;
        locsS[t * ST + 32] = 1.0f + 1e-6f;   // .set(1.0) then +SEARCH_EPS
        dS[t * ST + 0]     = 0.0f;

        float s0  = hrow[0]  / wrow[0];
        float s31 = hrow[31] / wrow[31];
        float dlft = 3.0f * s0  / (1.0f + expf(-udl[fg]));
        float drgt = 3.0f * s31 / (1.0f + expf(-udr[fg]));

        // derivatives[0..31] -> cS ; derivatives[32] kept in register
        cS[t * ST + 0] = dlft;
        float sp = s0, wp = wrow[0];
        for (int j = 1; j < KBINS; ++j) {
            float wj = wrow[j];
            float sj = hrow[j] / wj;
            float m1 = fminf(fabsf(sp), fabsf(sj));
            float m2 = 0.5f * (wj * sp + wp * sj) / (wp + wj);
            float mm = fminf(m1, m2);
            float sg = (float)((sp > 0.0f) - (sp < 0.0f)) + (float)((sj > 0.0f) - (sj < 0.0f));
            cS[t * ST + j] = mm * sg;
            sp = sj; wp = wj;
        }
        float dj = cS[t * ST + 0];
        for (int j = 0; j < KBINS; ++j) {
            float dj1 = (j < 31) ? cS[t * ST + j + 1] : drgt;
            float wj  = wrow[j];
            float sj  = hrow[j] / wj;
            aS[t * ST + j] = (dj + dj1 - 2.0f * sj) / (wj * wj);
            bS[t * ST + j] = (3.0f * sj - 2.0f * dj - dj1) / wj;
            dj = dj1;
        }
    }
    __syncthreads();

    // ---- Streaming evaluation: each wave = one row's 32-feature slice ----
    const int rowsPerBlock = NBATCH / ROWCHUNKS;     // 128
    const int rowsPerPass  = THREADS / FT;           // 16 (== #waves)
    const int rbase0 = blockIdx.y * rowsPerBlock;
    const int fl = lane;                              // FT == 32
    const int fglob = fbase + fl;
    const float* lr = &locsS[fl * ST];

    for (int p = 0; p < rowsPerBlock; p += rowsPerPass) {
        const int r = rbase0 + p + wave;
        const size_t gi = (size_t)r * Dfeat + fglob;
        float xv = x[gi];
        if (p + rowsPerPass < rowsPerBlock)
            __builtin_prefetch(&x[gi + (size_t)rowsPerPass * Dfeat], 0, 1);

        // branchless binary search: largest b in [0,31] with locs[b] <= xv
        int b = 0;
        if (xv >= lr[16])     b = 16;
        if (xv >= lr[b + 8])  b += 8;
        if (xv >= lr[b + 4])  b += 4;
        if (xv >= lr[b + 2])  b += 2;
        if (xv >= lr[b + 1])  b += 1;

        float s  = xv - lr[b];
        float ia = aS[fl * ST + b];
        float ib = bS[fl * ST + b];
        float ic = cS[fl * ST + b];
        float id = dS[fl * ST + b];

        float o = ((ia * s + ib) * s + ic) * s + id;
        o = fminf(fmaxf(o, 0.0f), 1.0f);
        out[gi] = o;

        float der = (3.0f * ia * s + 2.0f * ib) * s + ic;
        float l = logf(fabsf(der));
#pragma unroll
        for (int off = 16; off; off >>= 1) l += __shfl_xor(l, off, 32);
        if (lane == 0) atomicAdd(&lad[r], l);
    }
}

extern "C" void kernel_launch(void* const* d_in, const int* in_sizes, int n_in,
                              void* d_out, int out_size, void* d_ws, size_t ws_size,
                              hipStream_t stream) {
    (void)in_sizes; (void)n_in; (void)out_size; (void)d_ws; (void)ws_size;
    const float* x   = (const float*)d_in[0];
    const float* uw  = (const float*)d_in[1];
    const float* uh  = (const float*)d_in[2];
    const float* udl = (const float*)d_in[3];
    const float* udr = (const float*)d_in[4];
    float* out = (float*)d_out;
    float* lad = out + (size_t)NBATCH * Dfeat;

    zero_lad_kernel<<<(NBATCH + 255) / 256, 256, 0, stream>>>(lad);

    dim3 grid(Dfeat / FT, ROWCHUNKS);   // 32 x 32 = 1024 blocks
    spline_fused_kernel<<<grid, THREADS, 0, stream>>>(x, uw, uh, udl, udr, out, lad);
}